// AdaFocal_36180804502049
// MI455X (gfx1250) — compile-verified
//
#include <hip/hip_runtime.h>
#include <hip/hip_bf16.h>

typedef __attribute__((ext_vector_type(2))) float v2f;
typedef __attribute__((ext_vector_type(4))) float v4f;
typedef __attribute__((ext_vector_type(8))) float v8f;

#define MAX_BINS 32

// ---------------------------------------------------------------------------
// Phase 1 (fast path, C == 128): one row per wave32.
// 32 lanes * float4 = 128 cols = one contiguous 512B segment -> global_load_b128 (NT).
// ---------------------------------------------------------------------------
__global__ void adafocal_rows_c128(const float* __restrict__ inp,
                                   const int* __restrict__ target,
                                   const float* __restrict__ uppers,
                                   const float* __restrict__ gammas,
                                   float* __restrict__ partials,
                                   int N, int n_up) {
    __shared__ float s_up[MAX_BINS];
    __shared__ float s_gm[MAX_BINS + 1];
    __shared__ float s_wave[8];
    if (threadIdx.x < n_up)     s_up[threadIdx.x] = uppers[threadIdx.x];
    if (threadIdx.x < n_up + 1) s_gm[threadIdx.x] = gammas[threadIdx.x];
    __syncthreads();

    const int lane    = threadIdx.x & 31;
    const int waveBlk = threadIdx.x >> 5;
    const int gwave   = blockIdx.x * (blockDim.x >> 5) + waveBlk;
    const int nwaves  = gridDim.x * (blockDim.x >> 5);

    float acc = 0.0f;
    for (int row = gwave; row < N; row += nwaves) {
        const v4f* rp = reinterpret_cast<const v4f*>(inp + (size_t)row * 128);
        v4f v = __builtin_nontemporal_load(&rp[lane]);   // streaming read-once data

        // row max (lane-local then wave32 butterfly)
        float m = fmaxf(fmaxf(v.x, v.y), fmaxf(v.z, v.w));
        #pragma unroll
        for (int off = 16; off > 0; off >>= 1)
            m = fmaxf(m, __shfl_xor(m, off, 32));

        // sum of exp(x - m)
        float s = __expf(v.x - m) + __expf(v.y - m) + __expf(v.z - m) + __expf(v.w - m);
        #pragma unroll
        for (int off = 16; off > 0; off >>= 1)
            s += __shfl_xor(s, off, 32);

        // gather logit at target class (uniform per wave)
        int   t    = target[row];
        float comp = v.x;
        int   sub  = t & 3;
        if (sub == 1) comp = v.y;
        if (sub == 2) comp = v.z;
        if (sub == 3) comp = v.w;
        float xt = __shfl(comp, t >> 2, 32);

        float logpt = (xt - m) - __logf(s);
        float pt    = __expf(logpt);

        // searchsorted(side="right"): count of uppers <= pt
        int b = 0;
        for (int i = 0; i < n_up; ++i) b += (pt >= s_up[i]) ? 1 : 0;
        float g    = s_gm[b];
        float gs   = (g > 0.0f) ? 1.0f : ((g < 0.0f) ? -1.0f : 0.0f);
        float gmag = fabsf(g);
        float base = 1.0f - gs * pt + 1e-20f;
        acc += -__powf(base, gmag) * logpt;     // identical across all 32 lanes
    }

    if (lane == 0) s_wave[waveBlk] = acc;
    __syncthreads();
    if (threadIdx.x == 0) {
        float tsum = 0.0f;
        int nw = blockDim.x >> 5;
        for (int i = 0; i < nw; ++i) tsum += s_wave[i];
        partials[blockIdx.x] = tsum;
    }
}

// ---------------------------------------------------------------------------
// Phase 1 (generic C, multiple of 4): two passes over the row (2nd hits cache).
// ---------------------------------------------------------------------------
__global__ void adafocal_rows_gen(const float* __restrict__ inp,
                                  const int* __restrict__ target,
                                  const float* __restrict__ uppers,
                                  const float* __restrict__ gammas,
                                  float* __restrict__ partials,
                                  int N, int C, int n_up) {
    __shared__ float s_up[MAX_BINS];
    __shared__ float s_gm[MAX_BINS + 1];
    __shared__ float s_wave[8];
    if (threadIdx.x < n_up)     s_up[threadIdx.x] = uppers[threadIdx.x];
    if (threadIdx.x < n_up + 1) s_gm[threadIdx.x] = gammas[threadIdx.x];
    __syncthreads();

    const int lane    = threadIdx.x & 31;
    const int waveBlk = threadIdx.x >> 5;
    const int gwave   = blockIdx.x * (blockDim.x >> 5) + waveBlk;
    const int nwaves  = gridDim.x * (blockDim.x >> 5);
    const int C4      = C >> 2;

    float acc = 0.0f;
    for (int row = gwave; row < N; row += nwaves) {
        const v4f* rp = reinterpret_cast<const v4f*>(inp + (size_t)row * (size_t)C);
        float m = -3.402823466e+38f;
        for (int c4 = lane; c4 < C4; c4 += 32) {
            v4f v = rp[c4];
            m = fmaxf(m, fmaxf(fmaxf(v.x, v.y), fmaxf(v.z, v.w)));
        }
        #pragma unroll
        for (int off = 16; off > 0; off >>= 1)
            m = fmaxf(m, __shfl_xor(m, off, 32));

        float s = 0.0f;
        for (int c4 = lane; c4 < C4; c4 += 32) {
            v4f v = rp[c4];
            s += __expf(v.x - m) + __expf(v.y - m) + __expf(v.z - m) + __expf(v.w - m);
        }
        #pragma unroll
        for (int off = 16; off > 0; off >>= 1)
            s += __shfl_xor(s, off, 32);

        int   t  = target[row];
        float xt = (lane == 0) ? inp[(size_t)row * (size_t)C + t] : 0.0f;
        xt = __shfl(xt, 0, 32);

        float logpt = (xt - m) - __logf(s);
        float pt    = __expf(logpt);
        int b = 0;
        for (int i = 0; i < n_up; ++i) b += (pt >= s_up[i]) ? 1 : 0;
        float g    = s_gm[b];
        float gs   = (g > 0.0f) ? 1.0f : ((g < 0.0f) ? -1.0f : 0.0f);
        float gmag = fabsf(g);
        acc += -__powf(1.0f - gs * pt + 1e-20f, gmag) * logpt;
    }

    if (lane == 0) s_wave[waveBlk] = acc;
    __syncthreads();
    if (threadIdx.x == 0) {
        float tsum = 0.0f;
        int nw = blockDim.x >> 5;
        for (int i = 0; i < nw; ++i) tsum += s_wave[i];
        partials[blockIdx.x] = tsum;
    }
}

// ---------------------------------------------------------------------------
// Phase 2: single block reduces NP partials -> scalar.
// Final 256->1 collapse done with v_wmma_f32_16x16x4_f32 (ones B-matrix):
// D[m][n] = sum_k A[m][k], accumulated over 4 chunks of 64 values via C chain.
// A layout per ISA: lanes 0-15 hold M=0..15 with VGPR0=K0,VGPR1=K1; lanes 16-31 K2,K3.
// ---------------------------------------------------------------------------
__global__ void adafocal_reduce(const float* __restrict__ partials, int np,
                                float* __restrict__ out) {
    __shared__ float s[256];
    float a = 0.0f;
    for (int i = threadIdx.x; i < np; i += 256) a += partials[i];
    s[threadIdx.x] = a;
    __syncthreads();

    if (threadIdx.x < 32) {                 // wave-uniform guard: wave0 EXEC all-ones
        const int lane = threadIdx.x;
        const int m    = lane & 15;
        const int koff = (lane >> 4) << 1;  // 0 for lanes 0-15, 2 for lanes 16-31

        v2f ones; ones.x = 1.0f; ones.y = 1.0f;
        v8f c = {};                         // C starts at zero
        #pragma unroll
        for (int i = 0; i < 4; ++i) {
            int base = i * 64 + m * 4 + koff;
            v2f av; av.x = s[base]; av.y = s[base + 1];
            c = __builtin_amdgcn_wmma_f32_16x16x4_f32(
                    false, av, false, ones, (short)0, c, false, false);
        }
        // each column of D holds all 16 row-sums; lane holds 8 of them (M split at 16)
        float tsum = c[0] + c[1] + c[2] + c[3] + c[4] + c[5] + c[6] + c[7];
        tsum += __shfl_xor(tsum, 16, 32);
        if (lane == 0) out[0] = tsum;
    }
}

// ---------------------------------------------------------------------------
extern "C" void kernel_launch(void* const* d_in, const int* in_sizes, int n_in,
                              void* d_out, int out_size, void* d_ws, size_t ws_size,
                              hipStream_t stream) {
    const float* inp    = (const float*)d_in[0];
    const int*   target = (const int*)d_in[1];
    const float* uppers = (const float*)d_in[2];
    const float* gammas = (const float*)d_in[3];

    const int N    = in_sizes[1];
    const int C    = in_sizes[0] / N;
    const int n_up = in_sizes[2];

    float* partials = (float*)d_ws;
    const int NBLK = 4096;                 // 4096 partials = 16 KB of d_ws

    if (C == 128) {
        adafocal_rows_c128<<<NBLK, 256, 0, stream>>>(inp, target, uppers, gammas,
                                                     partials, N, n_up);
    } else {
        adafocal_rows_gen<<<NBLK, 256, 0, stream>>>(inp, target, uppers, gammas,
                                                    partials, N, C, n_up);
    }
    adafocal_reduce<<<1, 256, 0, stream>>>(partials, NBLK, (float*)d_out);
}